// MyHingeLoss_23888608101351
// MI455X (gfx1250) — compile-verified
//
#include <hip/hip_runtime.h>

// Poincare triplet hinge loss, MI455X (gfx1250, wave32).
// Memory-bound: ~193 MB traffic -> ~8.3 us floor @ 23.3 TB/s. Compute is free,
// so the five per-row dot products (un, vn, uv, tn, tv) are computed as
// diagonals of 16x16 Gram matrices with V_WMMA_F32_16X16X4_F32 over 16-row
// tiles. For f32 WMMA the A-layout of X and the B-layout of X^T coincide
// per-lane, so one float2 load per lane feeds both operands.

typedef float v2f __attribute__((ext_vector_type(2)));
typedef float v8f __attribute__((ext_vector_type(8)));

#define N_ROWS 262144
#define DIM 64
#define MARGIN_F 0.1f
#define EPS_F 1e-5f
#define TPB 256          // 8 wave32s per block
#define WAVES (TPB / 32)
#define BLOCKS (N_ROWS / (16 * WAVES))   // 2048; one 16-row tile per wave

__device__ __forceinline__ float vsel8(v8f a, int c) {
    // constant-index extracts only (avoid dynamic vector indexing -> scratch)
    float r = a[0];
    r = (c == 1) ? a[1] : r;
    r = (c == 2) ? a[2] : r;
    r = (c == 3) ? a[3] : r;
    r = (c == 4) ? a[4] : r;
    r = (c == 5) ? a[5] : r;
    r = (c == 6) ? a[6] : r;
    r = (c == 7) ? a[7] : r;
    return r;
}

__device__ __forceinline__ float acosh_clamped(float x) {
    // x already clamped >= 1 + EPS, so x*x - 1 > 0
    return logf(x + sqrtf(x * x - 1.0f));
}

__global__ void __launch_bounds__(TPB, 1)
poincare_hinge_kernel(const float* __restrict__ outv,   // output  [N, 64]
                      const float* __restrict__ tgt,    // target  [N, 64]
                      const int*   __restrict__ negIdx, // neg_idx [N]
                      float* __restrict__ partials)     // [BLOCKS]
{
    const int tid  = threadIdx.x;
    const int lane = tid & 31;
    const int wave = tid >> 5;
    const int tile = blockIdx.x * WAVES + wave;     // 16 rows per tile
    const int row  = (tile << 4) + (lane & 15);
    const int sub  = (lane >> 4) << 1;              // 0 (lanes 0-15) / 2 (lanes 16-31)

    const float* uRow = tgt  + (size_t)row * DIM;                    // target[i]
    const float* vRow = outv + (size_t)row * DIM;                    // output[i]
    const long   nIdx = (long)negIdx[row];
    const float* tRow = tgt  + (size_t)nIdx * DIM;                   // target[neg_idx[i]]

    v8f accUU = {};  // diag -> ||u||^2
    v8f accVV = {};  // diag -> ||v||^2
    v8f accUV = {};  // diag -> u.v
    v8f accTT = {};  // diag -> ||t||^2
    v8f accTV = {};  // diag -> t.v

#pragma unroll
    for (int k = 0; k < DIM / 4; ++k) {
        const int off = 4 * k + sub;
        v2f au = *(const v2f*)(uRow + off);
        v2f av = *(const v2f*)(vRow + off);
        v2f at = *(const v2f*)(tRow + off);
        accUU = __builtin_amdgcn_wmma_f32_16x16x4_f32(false, au, false, au, (short)0, accUU, false, false);
        accVV = __builtin_amdgcn_wmma_f32_16x16x4_f32(false, av, false, av, (short)0, accVV, false, false);
        accUV = __builtin_amdgcn_wmma_f32_16x16x4_f32(false, au, false, av, (short)0, accUV, false, false);
        accTT = __builtin_amdgcn_wmma_f32_16x16x4_f32(false, at, false, at, (short)0, accTT, false, false);
        accTV = __builtin_amdgcn_wmma_f32_16x16x4_f32(false, at, false, av, (short)0, accTV, false, false);
    }

    // Diagonal (i,i): i<8 -> lane i, comp i;  i>=8 -> lane i+16, comp i-8.
    const int  c      = lane & 7;
    const bool active = (lane < 8) || (lane >= 24);

    float un = vsel8(accUU, c);
    float vn = vsel8(accVV, c);
    float uv = vsel8(accUV, c);
    float tn = vsel8(accTT, c);
    float tv = vsel8(accTV, c);

    float sqp = un + vn - 2.0f * uv;            // ||u - v||^2
    float sqn = tn + vn - 2.0f * tv;            // ||t - v||^2
    float xp  = 1.0f + 2.0f * sqp / ((1.0f - un) * (1.0f - vn));
    float xn  = 1.0f + 2.0f * sqn / ((1.0f - tn) * (1.0f - vn));
    xp = fmaxf(xp, 1.0f + EPS_F);
    xn = fmaxf(xn, 1.0f + EPS_F);
    float dPos = acosh_clamped(xp);
    float dNeg = acosh_clamped(xn);
    float loss = fmaxf(MARGIN_F + dPos - dNeg, 0.0f);

    float contrib = active ? loss : 0.0f;       // select, no branch (EXEC stays full)
#pragma unroll
    for (int m = 16; m >= 1; m >>= 1)
        contrib += __shfl_xor(contrib, m, 32);

    __shared__ float wsum[WAVES];
    if (lane == 0) wsum[wave] = contrib;
    __syncthreads();
    if (tid == 0) {
        float s = 0.0f;
#pragma unroll
        for (int w = 0; w < WAVES; ++w) s += wsum[w];
        partials[blockIdx.x] = s;
    }
}

__global__ void __launch_bounds__(TPB, 1)
reduce_mean_kernel(const float* __restrict__ partials, float* __restrict__ out)
{
    __shared__ float sm[TPB];
    float s = 0.0f;
    for (int i = threadIdx.x; i < BLOCKS; i += TPB) s += partials[i];
    sm[threadIdx.x] = s;
    __syncthreads();
#pragma unroll
    for (int stride = TPB / 2; stride > 0; stride >>= 1) {
        if (threadIdx.x < stride) sm[threadIdx.x] += sm[threadIdx.x + stride];
        __syncthreads();
    }
    if (threadIdx.x == 0) out[0] = sm[0] * (1.0f / (float)N_ROWS);
}

extern "C" void kernel_launch(void* const* d_in, const int* in_sizes, int n_in,
                              void* d_out, int out_size, void* d_ws, size_t ws_size,
                              hipStream_t stream) {
    (void)in_sizes; (void)n_in; (void)out_size; (void)ws_size;
    const float* outv   = (const float*)d_in[0];  // "output" [N,64] f32
    const float* tgt    = (const float*)d_in[1];  // "target" [N,64] f32
    const int*   negIdx = (const int*)d_in[2];    // "neg_idx" [N] int32
    float* partials = (float*)d_ws;               // BLOCKS floats (8 KB)
    float* out = (float*)d_out;

    poincare_hinge_kernel<<<BLOCKS, TPB, 0, stream>>>(outv, tgt, negIdx, partials);
    reduce_mean_kernel<<<1, TPB, 0, stream>>>(partials, out);
}